// TopKRouter_77214922047953
// MI455X (gfx1250) — compile-verified
//
#include <hip/hip_runtime.h>
#include <hip/hip_bf16.h>

typedef __attribute__((ext_vector_type(2))) float v2f;
typedef __attribute__((ext_vector_type(8))) float v8f;

#define NUM_EXPERTS 64
#define DIM         1024
#define TOKENS      32768      // B*S = 4*8192
#define TOK_PER_BLOCK 128
#define THREADS     128
#define LDS_STRIDE  65         // 64 + 1 pad -> conflict-free column access

__global__ void router_zero_ws(float* __restrict__ ws) {
    int t = threadIdx.x;
    if (t < 129) ws[t] = 0.0f;
}

__global__ __launch_bounds__(THREADS)
void router_main(const float* __restrict__ x,     // [TOKENS, DIM]
                 const float* __restrict__ W,     // [NUM_EXPERTS, DIM]
                 float* __restrict__ out_rw,      // [TOKENS*2] router weights
                 float* __restrict__ out_idx,     // [TOKENS*2] expert indices (as float)
                 float* __restrict__ ws)          // [0,64) cnt  [64,128) probsum  [128] sumsq
{
    __shared__ float lds_logits[TOK_PER_BLOCK * LDS_STRIDE];
    __shared__ float lds_max[TOK_PER_BLOCK];
    __shared__ float lds_sum[TOK_PER_BLOCK];
    __shared__ float lds_cnt[NUM_EXPERTS];
    __shared__ float lds_prob[NUM_EXPERTS];
    __shared__ float lds_sq;

    const int tid  = threadIdx.x;
    const int wave = tid >> 5;
    const int lane = tid & 31;
    const int lrow = lane & 15;   // M (for A) / N (for B) within a 16-tile
    const int lhi  = lane >> 4;   // selects K pair {0,1} vs {2,3}

    if (tid < NUM_EXPERTS) { lds_cnt[tid] = 0.0f; lds_prob[tid] = 0.0f; }
    if (tid == 0) lds_sq = 0.0f;

    // ------------------------------------------------------------------
    // GEMM: 32 tokens (2 M-tiles) x 64 experts (4 N-tiles), K=1024 by 4
    // via V_WMMA_F32_16X16X4_F32 (pure f32 path; kernel is HBM-bound on x)
    // ------------------------------------------------------------------
    const int tok0 = blockIdx.x * TOK_PER_BLOCK + wave * 32;

    v8f acc[2][4] = {};   // zero-initialized accumulators

    // A layout (16x4 f32): lanes 0-15 hold M=lane, K={0,1}; lanes 16-31 M=lane-16, K={2,3}
    const float* xa0 = x + (size_t)(tok0 + lrow)      * DIM + 2 * lhi;
    const float* xa1 = x + (size_t)(tok0 + 16 + lrow) * DIM + 2 * lhi;
    // B layout (4x16 f32): lane holds column N=lane%16, K = 2*(lane/16)+{0,1}; B[k][n]=W[n][k]
    const float* wb0 = W + (size_t)( 0 + lrow) * DIM + 2 * lhi;
    const float* wb1 = W + (size_t)(16 + lrow) * DIM + 2 * lhi;
    const float* wb2 = W + (size_t)(32 + lrow) * DIM + 2 * lhi;
    const float* wb3 = W + (size_t)(48 + lrow) * DIM + 2 * lhi;

    #pragma unroll 2
    for (int k0 = 0; k0 < DIM; k0 += 4) {
        v2f a0 = *(const v2f*)(xa0 + k0);
        v2f a1 = *(const v2f*)(xa1 + k0);
        v2f b0 = *(const v2f*)(wb0 + k0);
        v2f b1 = *(const v2f*)(wb1 + k0);
        v2f b2 = *(const v2f*)(wb2 + k0);
        v2f b3 = *(const v2f*)(wb3 + k0);
        acc[0][0] = __builtin_amdgcn_wmma_f32_16x16x4_f32(false, a0, false, b0, (short)0, acc[0][0], false, false);
        acc[0][1] = __builtin_amdgcn_wmma_f32_16x16x4_f32(false, a0, false, b1, (short)0, acc[0][1], false, false);
        acc[0][2] = __builtin_amdgcn_wmma_f32_16x16x4_f32(false, a0, false, b2, (short)0, acc[0][2], false, false);
        acc[0][3] = __builtin_amdgcn_wmma_f32_16x16x4_f32(false, a0, false, b3, (short)0, acc[0][3], false, false);
        acc[1][0] = __builtin_amdgcn_wmma_f32_16x16x4_f32(false, a1, false, b0, (short)0, acc[1][0], false, false);
        acc[1][1] = __builtin_amdgcn_wmma_f32_16x16x4_f32(false, a1, false, b1, (short)0, acc[1][1], false, false);
        acc[1][2] = __builtin_amdgcn_wmma_f32_16x16x4_f32(false, a1, false, b2, (short)0, acc[1][2], false, false);
        acc[1][3] = __builtin_amdgcn_wmma_f32_16x16x4_f32(false, a1, false, b3, (short)0, acc[1][3], false, false);
    }

    // Scatter C tiles to LDS: C layout -> lane holds N=lane%16 (+16*nt), M=r+8*(lane/16)
    #pragma unroll
    for (int mt = 0; mt < 2; ++mt)
        #pragma unroll
        for (int nt = 0; nt < 4; ++nt)
            #pragma unroll
            for (int r = 0; r < 8; ++r) {
                int tloc = wave * 32 + 16 * mt + r + 8 * lhi;
                int e    = 16 * nt + lrow;
                lds_logits[tloc * LDS_STRIDE + e] = acc[mt][nt][r];
            }

    __syncthreads();

    // ------------------------------------------------------------------
    // Phase A: one thread per token -> top-2, router softmax, sumsq, cnt
    // ------------------------------------------------------------------
    {
        const float* row = lds_logits + tid * LDS_STRIDE;
        float m1 = -3.0e38f, m2 = -3.0e38f;
        int   i1 = 0, i2 = 0;
        float sumsq = 0.0f;
        #pragma unroll
        for (int e = 0; e < NUM_EXPERTS; ++e) {
            float l = row[e];
            sumsq += l * l;
            if (l > m1)      { m2 = m1; i2 = i1; m1 = l; i1 = e; }
            else if (l > m2) { m2 = l; i2 = e; }
        }
        float se = 0.0f;
        #pragma unroll
        for (int e = 0; e < NUM_EXPERTS; ++e)
            se += __expf(row[e] - m1);
        lds_max[tid] = m1;
        lds_sum[tid] = se;

        float e2  = __expf(m2 - m1);
        float inv = 1.0f / (1.0f + e2);
        size_t g = (size_t)blockIdx.x * TOK_PER_BLOCK + tid;
        out_rw[2 * g]     = inv;
        out_rw[2 * g + 1] = e2 * inv;
        out_idx[2 * g]     = (float)i1;
        out_idx[2 * g + 1] = (float)i2;

        atomicAdd(&lds_cnt[i1], 1.0f);

        // wave32 shuffle reduction of sumsq
        #pragma unroll
        for (int off = 16; off > 0; off >>= 1)
            sumsq += __shfl_xor(sumsq, off, 32);
        if (lane == 0) atomicAdd(&lds_sq, sumsq);
    }

    __syncthreads();

    // ------------------------------------------------------------------
    // Phase B: per-expert softmax prob sums; thread <-> (expert, half)
    // ------------------------------------------------------------------
    {
        int e     = tid & 63;
        int chunk = tid >> 6;                 // 0 or 1 (64 tokens each)
        float partial = 0.0f;
        int t0 = chunk * 64;
        #pragma unroll 4
        for (int t = t0; t < t0 + 64; ++t)
            partial += __expf(lds_logits[t * LDS_STRIDE + e] - lds_max[t]) / lds_sum[t];
        atomicAdd(&lds_prob[e], partial);
    }

    __syncthreads();

    if (tid < NUM_EXPERTS) {
        atomicAdd(&ws[tid], lds_cnt[tid]);
        atomicAdd(&ws[NUM_EXPERTS + tid], lds_prob[tid]);
    }
    if (tid == 0) atomicAdd(&ws[128], lds_sq);
}

__global__ void router_finalize(const float* __restrict__ ws,
                                float* __restrict__ out_scalar)
{
    __shared__ float red[NUM_EXPERTS];
    int t = threadIdx.x;
    red[t] = ws[t] * ws[NUM_EXPERTS + t];
    __syncthreads();
    if (t == 0) {
        float s = 0.0f;
        for (int i = 0; i < NUM_EXPERTS; ++i) s += red[i];
        const float T = 32768.0f;
        float aux = 64.0f * s / (T * T);
        float z   = ws[128] / (T * 64.0f) * 0.001f;
        out_scalar[0] = aux + z;
    }
}

extern "C" void kernel_launch(void* const* d_in, const int* in_sizes, int n_in,
                              void* d_out, int out_size, void* d_ws, size_t ws_size,
                              hipStream_t stream) {
    const float* x = (const float*)d_in[0];   // [4,8192,1024] f32
    const float* W = (const float*)d_in[1];   // [64,1024] f32
    float* out = (float*)d_out;               // rw[65536] | idx[65536] | loss[1]
    float* ws  = (float*)d_ws;                // 129 floats

    router_zero_ws<<<1, 256, 0, stream>>>(ws);
    router_main<<<TOKENS / TOK_PER_BLOCK, THREADS, 0, stream>>>(
        x, W, out, out + 2 * TOKENS, ws);
    router_finalize<<<1, NUM_EXPERTS, 0, stream>>>(ws, out + 4 * TOKENS);
}